// Quantizer_9672266350818
// MI455X (gfx1250) — compile-verified
//
#include <hip/hip_runtime.h>

typedef float v2f __attribute__((ext_vector_type(2)));
typedef float v8f __attribute__((ext_vector_type(8)));

#define N_EMBED 1024
#define EMBED_DIM 64
#define HW 4096
#define NBATCH 32
#define ROWS_PER_BLOCK 256      // 8 waves x 2 row-tiles x 16 rows
#define NBLOCKS ((NBATCH * HW) / ROWS_PER_BLOCK)   // 512
#define BETA 0.25f

// ---------------------------------------------------------------------------
// Kernel 1: per-codeword squared norms (codebook: 256 KB, L2-resident)
// ---------------------------------------------------------------------------
__global__ void k_embed_norms(const float* __restrict__ embed,
                              float* __restrict__ e2) {
    int c = blockIdx.x * blockDim.x + threadIdx.x;
    if (c < N_EMBED) {
        const float* row = embed + (size_t)c * EMBED_DIM;
        float s = 0.f;
#pragma unroll
        for (int d = 0; d < EMBED_DIM; ++d) {
            float v = row[d];
            s = fmaf(v, v, s);
        }
        e2[c] = s;
    }
}

// ---------------------------------------------------------------------------
// Kernel 2: argmin over codebook via fp32 WMMA + gather z_q + loss partials.
// Block = 256 threads = 8 wave32s; each wave owns TWO 16-row (hw) tiles so
// every B fragment (codeword tile) feeds two independent WMMA chains:
// halves L2 codebook traffic and doubles matrix-pipe ILP.
// score(row, col) = z.e - 0.5*||e||^2  (argmax score == argmin distance;
//  -0.5*||e||^2 is folded into the WMMA C accumulator seed).
// ---------------------------------------------------------------------------
__global__ __launch_bounds__(256) void k_vq(const float* __restrict__ x,
                                            const float* __restrict__ embed,
                                            const float* __restrict__ e2,
                                            float* __restrict__ zq_out,
                                            float* __restrict__ block_sums) {
    __shared__ int   sIdx[ROWS_PER_BLOCK];
    __shared__ float sWave[8];

    const int tid   = threadIdx.x;
    const int wave  = tid >> 5;
    const int lane  = tid & 31;
    const int col16 = lane & 15;   // N index of WMMA B/C/D fragments
    const int half  = lane >> 4;   // selects K pair (A/B) and M+8 (C/D)

    // 16 blocks per batch image: all 256 rows of a block share one b, contiguous hw
    const int b      = blockIdx.x >> 4;
    const int hwBase = (blockIdx.x & 15) * ROWS_PER_BLOCK;
    const float* xb  = x + (size_t)b * EMBED_DIM * HW;

    // ---- Phase 1: load two A fragments (2 x 16 rows x 64 dims, fp32) -------
    // A 16x4 layout: lanes 0-15 hold M=0..15; VGPR0 = K {0|2}, VGPR1 = K {1|3}
    const int hwRow0 = hwBase + wave * 32 + col16;   // tile 0, M = col16
    const int hwRow1 = hwRow0 + 16;                  // tile 1
    v2f a0[16], a1[16];
#pragma unroll
    for (int kk = 0; kk < 16; ++kk) {
        const int d0 = 4 * kk + 2 * half;
        a0[kk].x = xb[(size_t)d0 * HW + hwRow0];
        a0[kk].y = xb[(size_t)(d0 + 1) * HW + hwRow0];
        a1[kk].x = xb[(size_t)d0 * HW + hwRow1];
        a1[kk].y = xb[(size_t)(d0 + 1) * HW + hwRow1];
    }

    float best0[8], best1[8];
    int   bidx0[8], bidx1[8];
#pragma unroll
    for (int v = 0; v < 8; ++v) {
        best0[v] = -3.4e38f; bidx0[v] = 0;
        best1[v] = -3.4e38f; bidx1[v] = 0;
    }

    for (int ct = 0; ct < N_EMBED / 16; ++ct) {
        const int   colg = ct * 16 + col16;
        const float ce2  = e2[colg];
        const float* erow = embed + (size_t)colg * EMBED_DIM + 2 * half;

        if (ct + 1 < N_EMBED / 16)   // stream next codeword tile toward caches
            __builtin_prefetch(embed + (size_t)(colg + 16) * EMBED_DIM, 0, 3);

        // B 4x16 layout mirrors A: VGPR0 = K {0|2}, VGPR1 = K {1|3}, N = col16
        v2f bm[16];
#pragma unroll
        for (int kk = 0; kk < 16; ++kk)
            bm[kk] = *(const v2f*)(erow + 4 * kk);

        // Seed C with -0.5*||e_col||^2 (constant across M for a given N=lane)
        const float seed = -0.5f * ce2;
        v8f acc0, acc1;
#pragma unroll
        for (int v = 0; v < 8; ++v) { acc0[v] = seed; acc1[v] = seed; }

        // Two independent WMMA chains share each B fragment (2x reuse, 2x ILP)
#pragma unroll
        for (int kk = 0; kk < 16; ++kk) {
            acc0 = __builtin_amdgcn_wmma_f32_16x16x4_f32(
                false, a0[kk], false, bm[kk], (short)0, acc0, false, false);
            acc1 = __builtin_amdgcn_wmma_f32_16x16x4_f32(
                false, a1[kk], false, bm[kk], (short)0, acc1, false, false);
        }

        // D layout: VGPR v, lane -> (M = v + 8*half, N = col16)
#pragma unroll
        for (int v = 0; v < 8; ++v) {
            const float s0 = acc0[v];
            if (s0 > best0[v]) { best0[v] = s0; bidx0[v] = colg; }
            const float s1 = acc1[v];
            if (s1 > best1[v]) { best1[v] = s1; bidx1[v] = colg; }
        }
    }

    // Argmax across the 16 columns held by each half-wave (tie -> smaller idx,
    // matching jnp.argmin first-index semantics). XOR masks 8..1 stay in-half.
#pragma unroll
    for (int v = 0; v < 8; ++v) {
        float bv0 = best0[v]; int bi0 = bidx0[v];
        float bv1 = best1[v]; int bi1 = bidx1[v];
#pragma unroll
        for (int m = 8; m >= 1; m >>= 1) {
            const float ov0 = __shfl_xor(bv0, m, 32);
            const int   oi0 = __shfl_xor(bi0, m, 32);
            if (ov0 > bv0 || (ov0 == bv0 && oi0 < bi0)) { bv0 = ov0; bi0 = oi0; }
            const float ov1 = __shfl_xor(bv1, m, 32);
            const int   oi1 = __shfl_xor(bi1, m, 32);
            if (ov1 > bv1 || (ov1 == bv1 && oi1 < bi1)) { bv1 = ov1; bi1 = oi1; }
        }
        bidx0[v] = bi0;
        bidx1[v] = bi1;
    }
    if (col16 == 0) {
#pragma unroll
        for (int v = 0; v < 8; ++v) {
            sIdx[wave * 32 + half * 8 + v]      = bidx0[v];  // tile 0, m = v+8*half
            sIdx[wave * 32 + 16 + half * 8 + v] = bidx1[v];  // tile 1
        }
    }
    __syncthreads();

    // ---- Phase 2: gather codewords, emit NCHW z_q, accumulate loss --------
    // Thread t owns hw = hwBase + t; for each d, the 256 threads write one
    // fully contiguous 1 KB span of zq[b][d][...].
    const int    myIdx = sIdx[tid];
    const float* qrow  = embed + (size_t)myIdx * EMBED_DIM;
    const size_t off0  = (size_t)b * EMBED_DIM * HW + hwBase + tid;

    float lsum = 0.f;
#pragma unroll 8
    for (int d = 0; d < EMBED_DIM; ++d) {
        const float  q = qrow[d];
        const size_t o = off0 + (size_t)d * HW;
        const float zv = x[o];
        zq_out[o] = q;                            // straight-through fwd value
        const float diff = q - zv;
        lsum = fmaf(diff, diff, lsum);
    }

#pragma unroll
    for (int m = 16; m >= 1; m >>= 1) lsum += __shfl_xor(lsum, m, 32);
    if (lane == 0) sWave[wave] = lsum;
    __syncthreads();
    if (tid == 0) {
        float s = 0.f;
#pragma unroll
        for (int w = 0; w < 8; ++w) s += sWave[w];
        block_sums[blockIdx.x] = s;
    }
}

// ---------------------------------------------------------------------------
// Kernel 3: final loss reduction
// ---------------------------------------------------------------------------
__global__ void k_finish(const float* __restrict__ block_sums,
                         float* __restrict__ loss_out) {
    __shared__ float sh[256];
    float s = 0.f;
    for (int i = threadIdx.x; i < NBLOCKS; i += 256) s += block_sums[i];
    sh[threadIdx.x] = s;
    __syncthreads();
    if (threadIdx.x == 0) {
        float t = 0.f;
        for (int i = 0; i < 256; ++i) t += sh[i];
        const float nelem = (float)((size_t)NBATCH * EMBED_DIM * HW);
        loss_out[0] = (1.0f + BETA) * t / nelem;
    }
}

// ---------------------------------------------------------------------------
extern "C" void kernel_launch(void* const* d_in, const int* in_sizes, int n_in,
                              void* d_out, int out_size, void* d_ws, size_t ws_size,
                              hipStream_t stream) {
    (void)in_sizes; (void)n_in; (void)out_size; (void)ws_size;
    const float* x     = (const float*)d_in[0];
    const float* embed = (const float*)d_in[1];

    float* zq   = (float*)d_out;                                   // 8388608 floats
    float* loss = (float*)d_out + (size_t)NBATCH * EMBED_DIM * HW; // last element

    float* e2    = (float*)d_ws;          // [1024] codeword norms
    float* bsums = e2 + N_EMBED;          // [512] per-block loss partials

    k_embed_norms<<<4, 256, 0, stream>>>(embed, e2);
    k_vq<<<NBLOCKS, 256, 0, stream>>>(x, embed, e2, zq, bsums);
    k_finish<<<1, 256, 0, stream>>>(bsums, loss);
}